// HiCoEx_89807766159590
// MI455X (gfx1250) — compile-verified
//
#include <hip/hip_runtime.h>
#include <hip/hip_bf16.h>
#include <cstdint>

#define NODES   8192
#define FIN     256
#define FOUT    64
#define NEDGE   262144
#define LEAKY   0.2f
#define BN_EPSF 1e-5f

typedef __attribute__((ext_vector_type(2))) float v2f;
typedef __attribute__((ext_vector_type(8))) float v8f;

// ---------------------------------------------------------------------------
// Kernel 1: hp = h @ W  via V_WMMA_F32_16X16X4_F32.
// One wave -> one 16x16 tile of hp. Block = 128 threads = 4 waves covering the
// 64 output features; grid = 8192/16 row tiles. K=256 in steps of 4.
// A (16x4 f32): lane&15 = M; VGPR0 = K=(lane>=16?2:0), VGPR1 = K+1.
// B (4x16 f32): lane&15 = N; same K striping (ISA 7.12.2).
// C/D (16x16 f32): VGPR i -> row m0+i (+8 for lanes 16..31), col = lane&15.
// ---------------------------------------------------------------------------
__global__ __launch_bounds__(128)
void gat_gemm_wmma(const float* __restrict__ h, const float* __restrict__ W,
                   float* __restrict__ hp) {
  const int lane = threadIdx.x & 31;
  const int wave = threadIdx.x >> 5;
  const int m0   = blockIdx.x * 16;
  const int n0   = wave * 16;
  const int mi   = m0 + (lane & 15);
  const int ni   = n0 + (lane & 15);
  const int koff = (lane >> 4) << 1;           // 0 or 2

  const float* __restrict__ hrow = h + (size_t)mi * FIN;
  v8f c = {};
#pragma unroll 4
  for (int k0 = 0; k0 < FIN; k0 += 4) {
    v2f a, b;
    a.x = hrow[k0 + koff];
    a.y = hrow[k0 + koff + 1];
    b.x = W[(k0 + koff) * FOUT + ni];
    b.y = W[(k0 + koff + 1) * FOUT + ni];
    c = __builtin_amdgcn_wmma_f32_16x16x4_f32(false, a, false, b,
                                              (short)0, c, false, false);
  }
  const int rbase = m0 + ((lane >> 4) << 3);
#pragma unroll
  for (int i = 0; i < 8; ++i)
    hp[(size_t)(rbase + i) * FOUT + (n0 + (lane & 15))] = c[i];
}

// ---------------------------------------------------------------------------
// Kernel 2: zero-fill (adjacency bitmask) — graph-capture-safe memset.
// ---------------------------------------------------------------------------
__global__ void gat_zero_u32(uint32_t* __restrict__ p, int n) {
  int i = blockIdx.x * blockDim.x + threadIdx.x;
  if (i < n) p[i] = 0u;
}

// ---------------------------------------------------------------------------
// Kernel 3: f1 = hp@a1, f2 = hp@a2 (one thread per row).
// ---------------------------------------------------------------------------
__global__ void gat_f12(const float* __restrict__ hp, const float* __restrict__ avec,
                        float* __restrict__ f1, float* __restrict__ f2) {
  int i = blockIdx.x * blockDim.x + threadIdx.x;
  if (i >= NODES) return;
  const float* r = hp + (size_t)i * FOUT;
  float s1 = 0.f, s2 = 0.f;
#pragma unroll
  for (int f = 0; f < FOUT; ++f) {
    float v = r[f];
    s1 += v * avec[f];
    s2 += v * avec[FOUT + f];
  }
  f1[i] = s1;
  f2[i] = s2;
}

// ---------------------------------------------------------------------------
// Kernel 4: column sums of hp (for the all-NEG_FILL uniform-softmax fallback).
// One block per feature.
// ---------------------------------------------------------------------------
__global__ __launch_bounds__(256)
void gat_colsum(const float* __restrict__ hp, float* __restrict__ colsum) {
  __shared__ float sm[256];
  const int f = blockIdx.x;
  float s = 0.f;
  for (int i = threadIdx.x; i < NODES; i += 256) s += hp[(size_t)i * FOUT + f];
  sm[threadIdx.x] = s;
  __syncthreads();
  for (int o = 128; o > 0; o >>= 1) {
    if ((int)threadIdx.x < o) sm[threadIdx.x] += sm[threadIdx.x + o];
    __syncthreads();
  }
  if (threadIdx.x == 0) colsum[f] = sm[0];
}

// ---------------------------------------------------------------------------
// Kernel 5: adjacency bitmask build (dedups duplicate edges, like .set(True)).
// ---------------------------------------------------------------------------
__global__ void gat_adj(const int* __restrict__ ei, uint32_t* __restrict__ adj) {
  int e = blockIdx.x * blockDim.x + threadIdx.x;
  if (e >= NEDGE) return;
  int r = ei[e];
  int c = ei[NEDGE + e];
  atomicOr(&adj[(size_t)r * (NODES / 32) + (c >> 5)], 1u << (c & 31));
}

// ---------------------------------------------------------------------------
// Kernel 6: per-row online softmax over the bitmask + SpMM accumulate.
// One wave32 per row. Softmax state (m, s) is lane-uniform (all lanes walk the
// same bits); accumulation is lane-parallel: lane covers features f and f+32.
// ---------------------------------------------------------------------------
__global__ __launch_bounds__(256)
void gat_row_softmax_spmm(const uint32_t* __restrict__ adj,
                          const float* __restrict__ f1, const float* __restrict__ f2,
                          const float* __restrict__ hp, const float* __restrict__ colsum,
                          float* __restrict__ mrow, float* __restrict__ srow,
                          float* __restrict__ oraw) {
  const int row  = blockIdx.x * 8 + (threadIdx.x >> 5);
  const int lane = threadIdx.x & 31;
  if (row >= NODES) return;

  const uint32_t* __restrict__ w = adj + (size_t)row * (NODES / 32);
  const float f1i = f1[row];

  // Pass 1: uniform online softmax (running max m, running sum s).
  float m = -3.0e38f, s = 0.f;
  int   cnt = 0;
  for (int wi = 0; wi < NODES / 32; ++wi) {
    uint32_t bits = w[wi];
    while (bits) {
      int b = __ffs(bits) - 1;
      bits &= bits - 1;
      int j = (wi << 5) + b;
      float v = f1i + f2[j];
      v = v > 0.f ? v : LEAKY * v;
      if (v > m) { s = s * __expf(m - v) + 1.f; m = v; }
      else       { s += __expf(v - m); }
      ++cnt;
    }
  }

  float acc0 = 0.f, acc1 = 0.f;
  if (cnt == 0) {
    // softmax over all-NEG_FILL row -> uniform 1/N attention
    m = 0.f; s = 1.f;
    acc0 = colsum[lane]      * (1.f / (float)NODES);
    acc1 = colsum[lane + 32] * (1.f / (float)NODES);
  } else {
    const float inv = 1.f / s;
    for (int wi = 0; wi < NODES / 32; ++wi) {
      uint32_t bits = w[wi];
      while (bits) {
        int b = __ffs(bits) - 1;
        bits &= bits - 1;
        int j = (wi << 5) + b;
        float v = f1i + f2[j];
        v = v > 0.f ? v : LEAKY * v;
        float wgt = __expf(v - m) * inv;
        acc0 += wgt * hp[(size_t)j * FOUT + lane];
        acc1 += wgt * hp[(size_t)j * FOUT + 32 + lane];
      }
    }
  }
  if (lane == 0) { mrow[row] = m; srow[row] = s; }
  oraw[(size_t)row * FOUT + lane]      = acc0;
  oraw[(size_t)row * FOUT + 32 + lane] = acc1;
}

// ---------------------------------------------------------------------------
// Kernel 7: alpha[e] = att[row[e], col[e]] (rows with an out-edge are never
// the uniform-fallback case by construction).
// ---------------------------------------------------------------------------
__global__ void gat_alpha(const int* __restrict__ ei,
                          const float* __restrict__ f1, const float* __restrict__ f2,
                          const float* __restrict__ mrow, const float* __restrict__ srow,
                          float* __restrict__ alpha) {
  int e = blockIdx.x * blockDim.x + threadIdx.x;
  if (e >= NEDGE) return;
  int r = ei[e];
  int c = ei[NEDGE + e];
  float v = f1[r] + f2[c];
  v = v > 0.f ? v : LEAKY * v;
  alpha[e] = __expf(v - mrow[r]) / srow[r];
}

// ---------------------------------------------------------------------------
// Kernel 8: batch-norm column statistics (one block per feature).
// ---------------------------------------------------------------------------
__global__ __launch_bounds__(256)
void gat_bn_stats(const float* __restrict__ o, float* __restrict__ mean,
                  float* __restrict__ var) {
  __shared__ float s1[256];
  __shared__ float s2[256];
  const int f = blockIdx.x;
  float a = 0.f, b = 0.f;
  for (int i = threadIdx.x; i < NODES; i += 256) {
    float v = o[(size_t)i * FOUT + f];
    a += v;
    b += v * v;
  }
  s1[threadIdx.x] = a;
  s2[threadIdx.x] = b;
  __syncthreads();
  for (int off = 128; off > 0; off >>= 1) {
    if ((int)threadIdx.x < off) {
      s1[threadIdx.x] += s1[threadIdx.x + off];
      s2[threadIdx.x] += s2[threadIdx.x + off];
    }
    __syncthreads();
  }
  if (threadIdx.x == 0) {
    float mu = s1[0] * (1.f / (float)NODES);
    mean[f] = mu;
    var[f]  = s2[0] * (1.f / (float)NODES) - mu * mu;
  }
}

// ---------------------------------------------------------------------------
// Kernel 9: BN apply + bias -> d_out.
// ---------------------------------------------------------------------------
__global__ void gat_bn_apply(const float* __restrict__ o,
                             const float* __restrict__ mean, const float* __restrict__ var,
                             const float* __restrict__ gamma, const float* __restrict__ beta,
                             const float* __restrict__ bias, float* __restrict__ out) {
  int idx = blockIdx.x * blockDim.x + threadIdx.x;
  if (idx >= NODES * FOUT) return;
  int f = idx & (FOUT - 1);
  out[idx] = (o[idx] - mean[f]) * rsqrtf(var[f] + BN_EPSF) * gamma[f] + beta[f] + bias[f];
}

// ---------------------------------------------------------------------------
extern "C" void kernel_launch(void* const* d_in, const int* in_sizes, int n_in,
                              void* d_out, int out_size, void* d_ws, size_t ws_size,
                              hipStream_t stream) {
  const float* h     = (const float*)d_in[0];
  const int*   ei    = (const int*)d_in[1];
  const float* W     = (const float*)d_in[2];
  const float* avec  = (const float*)d_in[3];
  const float* bias  = (const float*)d_in[4];
  const float* gamma = (const float*)d_in[5];
  const float* beta  = (const float*)d_in[6];

  float* out   = (float*)d_out;            // [NODES*FOUT] normalized output
  float* alpha = out + NODES * FOUT;       // [NEDGE] per-edge attention

  // Workspace layout (floats), ~12.1 MB total.
  float*    ws     = (float*)d_ws;
  float*    hp     = ws;                         // 524288
  float*    f1     = hp + NODES * FOUT;          // 8192
  float*    f2     = f1 + NODES;                 // 8192
  float*    colsum = f2 + NODES;                 // 64
  float*    mrow   = colsum + FOUT;              // 8192
  float*    srow   = mrow + NODES;               // 8192
  float*    oraw   = srow + NODES;               // 524288
  float*    bnmean = oraw + NODES * FOUT;        // 64
  float*    bnvar  = bnmean + FOUT;              // 64
  uint32_t* adj    = (uint32_t*)(bnvar + FOUT);  // 8192*256 u32 = 8 MB bitmask

  const int adj_words = NODES * (NODES / 32);

  gat_zero_u32<<<(adj_words + 255) / 256, 256, 0, stream>>>(adj, adj_words);
  gat_gemm_wmma<<<NODES / 16, 128, 0, stream>>>(h, W, hp);
  gat_f12<<<(NODES + 255) / 256, 256, 0, stream>>>(hp, avec, f1, f2);
  gat_colsum<<<FOUT, 256, 0, stream>>>(hp, colsum);
  gat_adj<<<NEDGE / 256, 256, 0, stream>>>(ei, adj);
  gat_row_softmax_spmm<<<NODES / 8, 256, 0, stream>>>(adj, f1, f2, hp, colsum,
                                                      mrow, srow, oraw);
  gat_alpha<<<NEDGE / 256, 256, 0, stream>>>(ei, f1, f2, mrow, srow, alpha);
  gat_bn_stats<<<FOUT, 256, 0, stream>>>(oraw, bnmean, bnvar);
  gat_bn_apply<<<(NODES * FOUT + 255) / 256, 256, 0, stream>>>(oraw, bnmean, bnvar,
                                                               gamma, beta, bias, out);
}